// Net_58523224375950
// MI455X (gfx1250) — compile-verified
//
#include <hip/hip_runtime.h>
#include <hip/hip_bf16.h>

#define BATCH 1024
#define NSEL  4
#define SZc   2860
#define NPAIR 1378
#define NF    52

typedef __attribute__((ext_vector_type(2))) float v2f;
typedef __attribute__((ext_vector_type(8))) float v8f;

__device__ __forceinline__ v8f wmma4(v2f a, v2f b, v8f c) {
  // D = A(16x4) * B(4x16) + C, f32, wave32
  return __builtin_amdgcn_wmma_f32_16x16x4_f32(false, a, false, b, (short)0, c, false, false);
}
__device__ __forceinline__ float sp_(float x){ return (x > 20.f) ? x : log1pf(expf(x)); }
__device__ __forceinline__ float sg_(float x){ return 1.f / (1.f + expf(-x)); }

// blockDim.x must be 256
__device__ __forceinline__ float blk_sum(float v, float* red){
  int tid = threadIdx.x;
  red[tid] = v; __syncthreads();
  for (int o = 128; o > 0; o >>= 1){ if (tid < o) red[tid] += red[tid + o]; __syncthreads(); }
  float r = red[0]; __syncthreads();
  return r;
}
__device__ __forceinline__ float blk_max(float v, float* red){
  int tid = threadIdx.x;
  red[tid] = v; __syncthreads();
  for (int o = 128; o > 0; o >>= 1){ if (tid < o) red[tid] = fmaxf(red[tid], red[tid + o]); __syncthreads(); }
  float r = red[0]; __syncthreads();
  return r;
}

// ---------------- init: triu pair tables, last=1/NSEL, out=0 ----------------
__global__ void k_init(int* i0, int* i1, float* last, float* out4){
  int t = blockIdx.x * blockDim.x + threadIdx.x;
  if (t < NPAIR){
    int r = 0, start = 0;
    while (start + (NF - r) <= t){ start += NF - r; ++r; }
    i0[t] = r; i1[t] = r + (t - start);
  }
  if (t < SZc * NSEL) last[t] = 1.f / NSEL;
  if (t < BATCH * NSEL) out4[t] = 0.f;
}

// ---------------- scal = sel_scalar.sum(axis=-1): 262MB streaming pass ------
__global__ void k_scal(const float* __restrict__ ss, float* __restrict__ scal){
  __shared__ float red[256];
  int row = blockIdx.x;
  const float* p = ss + (size_t)row * SZc;
  float s = 0.f;
  for (int k = threadIdx.x; k < SZc; k += 256){
    if (k + 1024 < SZc) __builtin_prefetch(&p[k + 1024], 0, 1);
    s += p[k];
  }
  float t = blk_sum(s, red);
  if (threadIdx.x == 0) scal[row] = t;
}

// ---------------- generic WMMA f32 NT GEMM: C = act(A @ Bm^T + bias) --------
// A: M x K (lda), Bm: N x K (ldb). M % 16 == 0, K % 4 == 0, N arbitrary.
// act: 0 = identity, 1 = softplus. Sig (optional): sigmoid(pre-activation).
__global__ void k_gemm_nt(const float* __restrict__ A, int lda,
                          const float* __restrict__ Bm, int ldb,
                          const float* __restrict__ bias,
                          float* __restrict__ C, int ldc,
                          float* __restrict__ Sig,
                          int M, int N, int K, int act){
  const int lane = threadIdx.x & 31;
  const int wave = threadIdx.x >> 5;
  const int tn_tiles = (N + 15) >> 4;
  const int ntiles = (M >> 4) * tn_tiles;
  const int tile = blockIdx.x * (blockDim.x >> 5) + wave;
  if (tile >= ntiles) return;
  const int tm = tile / tn_tiles, tn = tile % tn_tiles;
  const int rc    = lane & 15;          // row of A tile == col of B tile
  const int khalf = (lane >> 4) << 1;   // 0 or 2
  const int half  = lane >> 4;

  const float* arow = A + (size_t)(tm * 16 + rc) * lda;
  const int n0 = tn * 16 + rc;
  const int nc = (n0 < N) ? n0 : (N - 1);
  const float bmask = (n0 < N) ? 1.f : 0.f;
  const float* brow = Bm + (size_t)nc * ldb;

  v8f c = {};
  for (int k0 = 0; k0 < K; k0 += 4){
    int kb = k0 + khalf;
    v2f a, b;
    a.x = arow[kb];          a.y = arow[kb + 1];
    b.x = brow[kb] * bmask;  b.y = brow[kb + 1] * bmask;
    c = wmma4(a, b, c);
  }
  for (int v = 0; v < 8; ++v){
    int m = tm * 16 + v + 8 * half;
    int n = tn * 16 + rc;
    if (n < N){
      float pre = c[v] + (bias ? bias[n] : 0.f);
      C[(size_t)m * ldc + n] = act ? sp_(pre) : pre;
      if (Sig) Sig[(size_t)m * ldc + n] = sg_(pre);
    }
  }
}

// ---------------- raw feature expansion + layernorm (fused, LDS) ------------
__global__ void k_raw_norm(const float* __restrict__ q, const float* __restrict__ dq,
                           const float* __restrict__ ddq, const float* __restrict__ out4,
                           const float* __restrict__ lng, const float* __restrict__ lnb,
                           const int* __restrict__ i0, const int* __restrict__ i1,
                           float* __restrict__ raw, float* __restrict__ normed){
  __shared__ float feat[NF];
  __shared__ float rl[SZc];
  __shared__ float red[256];
  int b = blockIdx.x, tid = threadIdx.x;
  if (tid < 16){ feat[tid] = q[b*16+tid]; feat[16+tid] = dq[b*16+tid]; feat[32+tid] = ddq[b*16+tid]; }
  if (tid < NSEL) feat[48+tid] = out4[b*NSEL+tid];
  __syncthreads();
  for (int t = tid; t < NPAIR; t += 256){
    int r = i0[t], cc = i1[t];
    float xi = feat[r], xj = feat[cc];
    rl[t] = (r == cc) ? xi : (xi + xj);
    rl[NPAIR + t] = xi * xj;
  }
  for (int j = tid; j < NF; j += 256){
    rl[2*NPAIR + j]      = sinf(feat[j]);
    rl[2*NPAIR + NF + j] = cosf(feat[j]);
  }
  __syncthreads();
  float s = 0.f, s2 = 0.f;
  for (int t = tid; t < SZc; t += 256){ float v = rl[t]; s += v; s2 += v*v; }
  float mu = blk_sum(s, red) * (1.f / SZc);
  float ms = blk_sum(s2, red) * (1.f / SZc);
  float inv = rsqrtf(fmaxf(ms - mu*mu, 0.f) + 1e-5f);
  size_t base = (size_t)b * SZc;
  for (int t = tid; t < SZc; t += 256){
    float v = rl[t];
    raw[base + t] = v;
    normed[base + t] = (v - mu) * inv * lng[t] + lnb[t];
  }
}

// ---------------- mse partials: sum((normed - xhat)^2) per sample -----------
__global__ void k_mse(const float* __restrict__ normed, const float* __restrict__ xhat,
                      float* __restrict__ partial){
  __shared__ float red[256];
  int b = blockIdx.x;
  size_t base = (size_t)b * SZc;
  float s = 0.f;
  for (int o = threadIdx.x; o < SZc; o += 256){
    float d = normed[base + o] - xhat[base + o]; s += d*d;
  }
  float t = blk_sum(s, red);
  if (threadIdx.x == 0) partial[b] = t;
}

__global__ void k_reduce(const float* __restrict__ src, int n, float scale,
                         float* __restrict__ dst){
  __shared__ float red[256];
  float s = 0.f;
  for (int i = threadIdx.x; i < n; i += 256) s += src[i];
  float t = blk_sum(s, red);
  if (threadIdx.x == 0) dst[0] = t * scale;
}

// ---------------- per-sample Jacobian Frobenius: ||W3 D2 W2 D1 W1||_F^2 -----
// Uses G = W1 W1^T. Phase 1: M(48x128) = (W3*s2cols) @ (W2*s1cols) via WMMA.
// Phase 2: sum over (M^T M) elementwise G, via WMMA. One workgroup per sample.
__global__ void k_jac(const float* __restrict__ s1g, const float* __restrict__ s2g,
                      const float* __restrict__ W2, const float* __restrict__ W3,
                      const float* __restrict__ G, float* __restrict__ jacF){
  __shared__ float s1[128], s2[128];
  __shared__ float Ml[48 * 128];
  __shared__ float red[256];
  int b = blockIdx.x, tid = threadIdx.x;
  int lane = tid & 31, wave = tid >> 5;
  if (tid < 128){ s1[tid] = s1g[b*128 + tid]; s2[tid] = s2g[b*128 + tid]; }
  __syncthreads();
  const int rc = lane & 15, khalf = (lane >> 4) << 1, half = lane >> 4;
  // Phase 1: 24 tiles (3 x 8)
  for (int t = wave; t < 24; t += 8){
    int tm = t / 8, tn = t % 8;
    int n = tn * 16 + rc;
    v8f c = {};
    for (int k0 = 0; k0 < 128; k0 += 4){
      int kb = k0 + khalf;
      v2f a, bb;
      a.x  = W3[(tm*16 + rc)*128 + kb]     * s2[kb];
      a.y  = W3[(tm*16 + rc)*128 + kb + 1] * s2[kb + 1];
      bb.x = W2[kb*128 + n]       * s1[n];
      bb.y = W2[(kb + 1)*128 + n] * s1[n];
      c = wmma4(a, bb, c);
    }
    for (int v = 0; v < 8; ++v) Ml[(tm*16 + v + 8*half)*128 + n] = c[v];
  }
  __syncthreads();
  // Phase 2: P = M^T M (128x128), acc += P . G
  float acc = 0.f;
  for (int t = wave; t < 64; t += 8){
    int ta = t / 8, tb = t % 8;
    v8f c = {};
    for (int k0 = 0; k0 < 48; k0 += 4){
      int kb = k0 + khalf;
      v2f a, bb;
      a.x  = Ml[kb*128 + ta*16 + rc];  a.y  = Ml[(kb+1)*128 + ta*16 + rc];
      bb.x = Ml[kb*128 + tb*16 + rc];  bb.y = Ml[(kb+1)*128 + tb*16 + rc];
      c = wmma4(a, bb, c);
    }
    for (int v = 0; v < 8; ++v){
      int m = ta*16 + v + 8*half, n = tb*16 + rc;
      acc += c[v] * G[m*128 + n];
    }
  }
  float tot = blk_sum(acc, red);
  if (tid == 0) jacF[b] = tot;
}

// ---------------- zz = z^T z, zaza = |z|^T |z| (48x48, K=1024) via WMMA -----
__global__ void k_zz(const float* __restrict__ z, float* __restrict__ zz,
                     float* __restrict__ zaza){
  int tid = threadIdx.x, lane = tid & 31, wave = tid >> 5;  // 9 waves
  int tm = wave / 3, tn = wave % 3;
  const int rc = lane & 15, khalf = (lane >> 4) << 1, half = lane >> 4;
  v8f c0 = {}, c1 = {};
  for (int k0 = 0; k0 < BATCH; k0 += 4){
    int kb = k0 + khalf;
    v2f a, b, aa, ab;
    a.x = z[kb*48 + tm*16 + rc];       a.y = z[(kb+1)*48 + tm*16 + rc];
    b.x = z[kb*48 + tn*16 + rc];       b.y = z[(kb+1)*48 + tn*16 + rc];
    c0 = wmma4(a, b, c0);
    aa.x = fabsf(a.x); aa.y = fabsf(a.y);
    ab.x = fabsf(b.x); ab.y = fabsf(b.y);
    c1 = wmma4(aa, ab, c1);
  }
  for (int v = 0; v < 8; ++v){
    int m = tm*16 + v + 8*half, n = tn*16 + rc;
    zz[m*48 + n]   = c0[v];
    zaza[m*48 + n] = c1[v];
  }
}

// ---------------- selector head (all tiny ops fused, one workgroup) ---------
__global__ void k_head(const float* __restrict__ zz, const float* __restrict__ zaza,
                       const float* __restrict__ q1w, const float* __restrict__ q1b,
                       const float* __restrict__ selq, const float* __restrict__ sl1,
                       const float* __restrict__ lg1, const float* __restrict__ lb1,
                       const float* __restrict__ sl2, const float* __restrict__ lg2,
                       const float* __restrict__ lb2, const float* __restrict__ sonoff,
                       const float* __restrict__ scal,
                       float* __restrict__ last, float* __restrict__ scaling,
                       float* __restrict__ gout){
  __shared__ float buf[4 * SZc];   // phase A: gsa[0..2304) attn[2304..4608); phase B: pre2/out_sm
  __shared__ float sub[4 * 48];
  __shared__ float h1s[4 * 64];
  __shared__ float red[256];
  __shared__ float onoff[4];
  __shared__ float gces[16];
  int tid = threadIdx.x;
  float* gsa  = buf;
  float* attn = buf + 2304;

  for (int t = tid; t < 2304; t += 256) gsa[t] = zz[t] / fmaxf(zaza[t], 1e-5f);
  __syncthreads();
  for (int t = tid; t < 2304; t += 256){
    int r = t / 48, c = t % 48;
    float s = q1b[c];
    const float* wr = q1w + c * 48;
    for (int l = 0; l < 48; ++l) s += gsa[r*48 + l] * wr[l];
    attn[t] = s;
  }
  __syncthreads();
  for (int t = tid; t < 4*48; t += 256){
    int h = t / 48, r = t % 48;
    float s = 0.f;
    for (int l = 0; l < 48; ++l) s += selq[h*48 + l] * attn[r*48 + l];
    sub[h*48 + r] = s;
  }
  __syncthreads();
  { // t1[h][k] = sel_l1[h,k,:] . sub[h,:]   (4*64 == blockDim)
    int h = tid / 64, k = tid % 64;
    float s = 0.f;
    const float* wr = sl1 + (h*64 + k) * 48;
    for (int l = 0; l < 48; ++l) s += wr[l] * sub[h*48 + l];
    h1s[tid] = s;
  }
  __syncthreads();
  if (tid < 4){ // GN(64) + softplus per head
    int h = tid;
    float mu = 0.f; for (int k = 0; k < 64; ++k) mu += h1s[h*64 + k]; mu *= (1.f/64.f);
    float var = 0.f; for (int k = 0; k < 64; ++k){ float d = h1s[h*64 + k] - mu; var += d*d; } var *= (1.f/64.f);
    float inv = rsqrtf(var + 1e-5f);
    for (int k = 0; k < 64; ++k){
      float v = (h1s[h*64 + k] - mu) * inv * lg1[h*64 + k] + lb1[h*64 + k];
      h1s[h*64 + k] = sp_(v);
    }
  }
  __syncthreads();
  // pre2[h][o] (overwrites gsa/attn region)
  for (int t = tid; t < 4*SZc; t += 256){
    int h = t / SZc, o = t % SZc;
    const float* wr = sl2 + (size_t)(h*SZc + o) * 64;
    float s = 0.f;
    for (int k = 0; k < 64; ++k) s += wr[k] * h1s[h*64 + k];
    buf[t] = s;
  }
  __syncthreads();
  // per head: GN(SZ) then softmax(SZ)
  for (int h = 0; h < 4; ++h){
    float* p = buf + h * SZc;
    float s = 0.f; for (int o = tid; o < SZc; o += 256) s += p[o];
    float mu = blk_sum(s, red) * (1.f / SZc);
    s = 0.f; for (int o = tid; o < SZc; o += 256){ float d = p[o] - mu; s += d*d; }
    float inv = rsqrtf(blk_sum(s, red) * (1.f / SZc) + 1e-5f);
    float mx = -3.0e38f;
    for (int o = tid; o < SZc; o += 256){
      float v = (p[o] - mu) * inv * lg2[h*SZc + o] + lb2[h*SZc + o];
      p[o] = v; mx = fmaxf(mx, v);
    }
    mx = blk_max(mx, red);
    s = 0.f;
    for (int o = tid; o < SZc; o += 256){ float e = expf(p[o] - mx); p[o] = e; s += e; }
    float isum = 1.f / blk_sum(s, red);
    for (int o = tid; o < SZc; o += 256) p[o] *= isum;   // out_sm
    __syncthreads();
  }
  // onoff[h] = sigmoid(sel_onoff[h] . [inp_last | out_sm[h]])
  for (int h = 0; h < 4; ++h){
    const float* so = sonoff + (size_t)h * 2 * SZc;
    float s = 0.f;
    for (int o = tid; o < SZc; o += 256){
      float il = last[o*4+0] * last[o*4+1] * last[o*4+2] * last[o*4+3];
      s += so[o] * il + so[SZc + o] * buf[h*SZc + o];
    }
    float tot = blk_sum(s, red);
    if (tid == 0) onoff[h] = sg_(tot);
  }
  __syncthreads();
  // scaling (scaled^T, with where) and new -> last
  for (int o = tid; o < SZc; o += 256){
    for (int h = 0; h < 4; ++h){
      float on = onoff[h] * buf[h*SZc + o];
      float sc = on * scal[h*SZc + o];
      sc = (fabsf(sc) <= 1e-14f) ? 1e-14f : sc;
      scaling[o*4 + h] = sc;
      last[o*4 + h] = fmaxf(on, 1e-14f);
    }
  }
  __syncthreads();
  // gce[a][b] = -sum_s new[s,a]*log(new[s,b]+1e-5); then g
  float part[16];
  for (int e = 0; e < 16; ++e) part[e] = 0.f;
  for (int o = tid; o < SZc; o += 256){
    float nv[4], lv[4];
    for (int h = 0; h < 4; ++h){ nv[h] = last[o*4 + h]; lv[h] = logf(nv[h] + 1e-5f); }
    for (int a = 0; a < 4; ++a)
      for (int b = 0; b < 4; ++b) part[a*4 + b] -= nv[a] * lv[b];
  }
  for (int e = 0; e < 16; ++e){
    float t = blk_sum(part[e], red);
    if (tid == 0) gces[e] = t;
  }
  __syncthreads();
  if (tid == 0){
    float dsum = gces[0] + gces[5] + gces[10] + gces[15];
    float asum = 0.f; for (int e = 0; e < 16; ++e) asum += gces[e];
    gout[0] = 0.001f * (dsum * 0.25f) - (asum - dsum) * (1.f / 16.f);
  }
}

// ---------------- out[b][h] = raw[b,:] . scaling[:,h] -----------------------
__global__ void k_out(const float* __restrict__ raw, const float* __restrict__ scaling,
                      float* __restrict__ out4){
  __shared__ float red[256];
  int b = blockIdx.x, tid = threadIdx.x;
  const float* r = raw + (size_t)b * SZc;
  float a0 = 0, a1 = 0, a2 = 0, a3 = 0;
  for (int o = tid; o < SZc; o += 256){
    float rv = r[o];
    a0 += rv * scaling[o*4+0]; a1 += rv * scaling[o*4+1];
    a2 += rv * scaling[o*4+2]; a3 += rv * scaling[o*4+3];
  }
  float t;
  t = blk_sum(a0, red); if (tid == 0) out4[b*4+0] = t;
  t = blk_sum(a1, red); if (tid == 0) out4[b*4+1] = t;
  t = blk_sum(a2, red); if (tid == 0) out4[b*4+2] = t;
  t = blk_sum(a3, red); if (tid == 0) out4[b*4+3] = t;
}

__global__ void k_final(const float* __restrict__ out4, const float* __restrict__ scalars,
                        float* __restrict__ dout){
  int b = blockIdx.x * blockDim.x + threadIdx.x;
  if (b < BATCH) dout[b] = out4[b*4+0] + out4[b*4+1] + out4[b*4+2] + out4[b*4+3];
  if (b == 0)
    dout[BATCH] = scalars[0] + scalars[1] + scalars[2] + scalars[3] + scalars[4] + scalars[5];
}

extern "C" void kernel_launch(void* const* d_in, const int* in_sizes, int n_in,
                              void* d_out, int out_size, void* d_ws, size_t ws_size,
                              hipStream_t stream) {
  const float* q      = (const float*)d_in[0];
  const float* dq     = (const float*)d_in[1];
  const float* ddq    = (const float*)d_in[2];
  const float* ln_g   = (const float*)d_in[3];
  const float* ln_b   = (const float*)d_in[4];
  const float* ew1    = (const float*)d_in[5];
  const float* eb1    = (const float*)d_in[6];
  const float* ew2    = (const float*)d_in[7];
  const float* eb2    = (const float*)d_in[8];
  const float* ew3    = (const float*)d_in[9];
  const float* eb3    = (const float*)d_in[10];
  const float* dw1    = (const float*)d_in[11];
  const float* db1    = (const float*)d_in[12];
  const float* dw2    = (const float*)d_in[13];
  const float* db2    = (const float*)d_in[14];
  const float* dw3    = (const float*)d_in[15];
  const float* db3    = (const float*)d_in[16];
  const float* q1w    = (const float*)d_in[17];
  const float* q1b    = (const float*)d_in[18];
  const float* selq   = (const float*)d_in[19];
  const float* sl1    = (const float*)d_in[20];
  const float* lg1    = (const float*)d_in[21];
  const float* lb1    = (const float*)d_in[22];
  const float* sl2    = (const float*)d_in[23];
  const float* lg2    = (const float*)d_in[24];
  const float* lb2    = (const float*)d_in[25];
  const float* sscal  = (const float*)d_in[26];
  const float* sonoff = (const float*)d_in[27];

  char* wsb = (char*)d_ws;
  size_t off = 0;
  auto alloc = [&](size_t nfloats) -> void* {
    void* p = wsb + off; off += nfloats * sizeof(float); return p;
  };
  int*   i0      = (int*)  alloc(NPAIR);
  int*   i1      = (int*)  alloc(NPAIR);
  float* last    = (float*)alloc((size_t)SZc * NSEL);
  float* out4    = (float*)alloc((size_t)BATCH * NSEL);
  float* raw     = (float*)alloc((size_t)BATCH * SZc);
  float* normed  = (float*)alloc((size_t)BATCH * SZc);
  float* xhat    = (float*)alloc((size_t)BATCH * SZc);
  float* h1e     = (float*)alloc((size_t)BATCH * 128);
  float* s1      = (float*)alloc((size_t)BATCH * 128);
  float* h2e     = (float*)alloc((size_t)BATCH * 128);
  float* s2      = (float*)alloc((size_t)BATCH * 128);
  float* zbuf    = (float*)alloc((size_t)BATCH * 48);
  float* hd1     = (float*)alloc((size_t)BATCH * 128);
  float* hd2     = (float*)alloc((size_t)BATCH * 128);
  float* G       = (float*)alloc(128 * 128);
  float* zzb     = (float*)alloc(48 * 48);
  float* zazab   = (float*)alloc(48 * 48);
  float* scalbuf = (float*)alloc((size_t)2 * NSEL * SZc);
  float* jacF    = (float*)alloc(BATCH);
  float* msep    = (float*)alloc(BATCH);
  float* scaling = (float*)alloc((size_t)SZc * NSEL);
  float* scalars = (float*)alloc(8);

  auto gemm = [&](const float* A, int lda, const float* Bm, int ldb, const float* bias,
                  float* C, int ldc, float* Sig, int M, int N, int K, int act){
    int tiles = (M / 16) * ((N + 15) / 16);
    k_gemm_nt<<<(tiles + 7) / 8, 256, 0, stream>>>(A, lda, Bm, ldb, bias, C, ldc, Sig,
                                                   M, N, K, act);
  };

  k_init<<<45, 256, 0, stream>>>(i0, i1, last, out4);
  k_scal<<<2 * NSEL * SZc, 256, 0, stream>>>(sscal, scalbuf);
  // G = enc_w1 @ enc_w1^T  (128x128, K=2860), once
  gemm(ew1, SZc, ew1, SZc, nullptr, G, 128, nullptr, 128, 128, SZc, 0);

  for (int i = 0; i < 2; ++i){
    k_raw_norm<<<BATCH, 256, 0, stream>>>(q, dq, ddq, out4, ln_g, ln_b, i0, i1, raw, normed);
    // encoder (store sigmoid(pre) for Jacobian)
    gemm(normed, SZc, ew1, SZc, eb1, h1e, 128, s1, BATCH, 128, SZc, 1);
    gemm(h1e, 128, ew2, 128, eb2, h2e, 128, s2, BATCH, 128, 128, 1);
    gemm(h2e, 128, ew3, 128, eb3, zbuf, 48, nullptr, BATCH, 48, 128, 0);
    // decoder
    gemm(zbuf, 48, dw1, 48, db1, hd1, 128, nullptr, BATCH, 128, 48, 1);
    gemm(hd1, 128, dw2, 128, db2, hd2, 128, nullptr, BATCH, 128, 128, 1);
    gemm(hd2, 128, dw3, 128, db3, xhat, SZc, nullptr, BATCH, SZc, 128, 0);
    // cst terms
    k_mse<<<BATCH, 256, 0, stream>>>(normed, xhat, msep);
    k_reduce<<<1, 256, 0, stream>>>(msep, BATCH, 1.f / ((float)BATCH * (float)SZc),
                                    scalars + 3*i + 1);
    k_jac<<<BATCH, 256, 0, stream>>>(s1, s2, ew2, ew3, G, jacF);
    k_reduce<<<1, 256, 0, stream>>>(jacF, BATCH, 1.f / ((float)BATCH * 48.f * (float)SZc),
                                    scalars + 3*i + 2);
    // correlation + selector head
    k_zz<<<1, 288, 0, stream>>>(zbuf, zzb, zazab);
    k_head<<<1, 256, 0, stream>>>(zzb, zazab,
        q1w + (size_t)i*48*48, q1b + (size_t)i*48,
        selq + (size_t)i*4*48, sl1 + (size_t)i*4*64*48,
        lg1 + (size_t)i*4*64, lb1 + (size_t)i*4*64,
        sl2 + (size_t)i*4*SZc*64, lg2 + (size_t)i*4*SZc, lb2 + (size_t)i*4*SZc,
        sonoff + (size_t)i*4*2*SZc, scalbuf + (size_t)i*4*SZc,
        last, scaling, scalars + 3*i);
    k_out<<<BATCH, 256, 0, stream>>>(raw, scaling, out4);
  }
  k_final<<<4, 256, 0, stream>>>(out4, scalars, (float*)d_out);
}